// Decoder_73254962200709
// MI455X (gfx1250) — compile-verified
//
#include <hip/hip_runtime.h>
#include <hip/hip_bf16.h>

// LSTM decoder, P=81 steps, E=256, H=4096.
//   z_t = x_t@W[:E] + h_t@(W[E:]+U) + b ; gates i,f,g,o ; linear candidate/output.
// Phase 1 (once):  Zx[81,16K] = h_enc @ W[:E] + b   -> V_WMMA_F32_16X16X4_F32 GEMM
//                  Wsum       = W[E:] + U            (if workspace permits: halves stream)
// Phase 2 (81x):   split-K GEMV h@Wsum -> partials; fused gate kernel reduces,
//                  updates (h,c), writes orders row.
//
// L2 pinning (MI455X: 192MB global L2, 23.3 TB/s HBM): Wsum = 268MB does not fit, so
// uniform loads re-stream 268MB/step from HBM (21.7 GB total, ~0.93ms). Instead we keep
// K-chunks 0..19 (~168MB) resident with RT loads and stream chunks 20..31 (~100MB) with
// TH=NT loads so they cannot evict the resident slice: ~100MB/step HBM => ~0.35ms.

typedef float v2f __attribute__((ext_vector_type(2)));
typedef float v4f __attribute__((ext_vector_type(4)));
typedef float v8f __attribute__((ext_vector_type(8)));

static constexpr int P      = 81;
static constexpr int E      = 256;
static constexpr int H      = 4096;
static constexpr int FOURH  = 4 * H;            // 16384
static constexpr int KSPLIT = 32;               // k-chunks per GEMV (8.39MB each, fused)
static constexpr int KCH    = H / KSPLIT;       // 128
static constexpr int CBLK   = 16;               // column blocks per GEMV
static constexpr int CPB    = FOURH / CBLK;     // 1024 cols per block (4/thread, float4)
static constexpr int KEEP_FUSED = 20;           // resident chunks: 20*8.39MB ~ 168MB < 192MB L2
static constexpr int KEEP_FB    = 10;           // fallback reads W+U (16.8MB/chunk): 10 ~ 168MB

// ---------------- Phase 1a: Zx = h_enc @ W[:E] + b via WMMA f32 16x16x4 -------------
// One wave per 16x16 output tile. grid = (1024 nTiles, 6 mTiles), block = 32 (one wave).
__global__ __launch_bounds__(32) void zx_wmma_kernel(
    const float* __restrict__ henc,   // [P, E]
    const float* __restrict__ W,      // [E+H, 4H] (top E rows used)
    const float* __restrict__ bias,   // [4H]
    float* __restrict__ Zx)           // [P, 4H]
{
  const int lane = threadIdx.x;
  const int half = lane >> 4;         // 0: lanes 0-15, 1: lanes 16-31
  const int ln   = lane & 15;
  const int col  = blockIdx.x * 16 + ln;         // N index
  const int m    = blockIdx.y * 16 + ln;         // M index (row of h_enc)
  const int mc   = (m < P) ? m : (P - 1);        // clamped; mask via multiply (no EXEC div)
  const float msk = (m < P) ? 1.0f : 0.0f;

  const float bv = bias[col];
  v8f acc;
  #pragma unroll
  for (int j = 0; j < 8; ++j) acc[j] = bv;       // C added once along the WMMA chain

  const float* arow = henc + (size_t)mc * E;
  #pragma unroll 4
  for (int k0 = 0; k0 < E; k0 += 4) {
    const int ka = k0 + 2 * half;                // A: lane half picks K {0,1} / {2,3}
    v2f a; a.x = arow[ka] * msk; a.y = arow[ka + 1] * msk;
    v2f b;                                       // B: same half split on K rows
    b.x = W[(size_t)ka * FOURH + col];
    b.y = W[(size_t)(ka + 1) * FOURH + col];
    acc = __builtin_amdgcn_wmma_f32_16x16x4_f32(
        /*neg_a=*/false, a, /*neg_b=*/false, b,
        /*c_mod=*/(short)0, acc, /*reuse_a=*/false, /*reuse_b=*/false);
  }

  #pragma unroll
  for (int j = 0; j < 8; ++j) {                  // D: VGPR j -> M = j + 8*half
    const int mr = blockIdx.y * 16 + j + 8 * half;
    if (mr < P) Zx[(size_t)mr * FOURH + col] = acc[j];
  }
}

// ---------------- Phase 1b: Wsum = W[E:] + U (grid-stride, float4) ------------------
// One-shot reads -> non-temporal loads. Stores: resident head RT (seed L2),
// streaming tail NT (don't pollute).
__global__ __launch_bounds__(256) void wsum_kernel(
    const float* __restrict__ W, const float* __restrict__ U,
    float* __restrict__ Wsum, long n4, long n4keep)
{
  const v4f* Wr = (const v4f*)(W + (size_t)E * FOURH);
  const v4f* Ur = (const v4f*)U;
  v4f* S = (v4f*)Wsum;
  long i = (long)blockIdx.x * blockDim.x + threadIdx.x;
  const long stride = (long)gridDim.x * blockDim.x;
  for (; i < n4; i += stride) {
    const v4f a = __builtin_nontemporal_load(Wr + i);
    const v4f b = __builtin_nontemporal_load(Ur + i);
    const v4f s = a + b;
    if (i < n4keep) S[i] = s;
    else            __builtin_nontemporal_store(s, S + i);
  }
}

// ---------------- Init: h=h0, c=0, orders[0]=h0 -------------------------------------
__global__ __launch_bounds__(256) void init_kernel(
    const float* __restrict__ h0, float* __restrict__ hbuf,
    float* __restrict__ cbuf, float* __restrict__ out0)
{
  const int j = blockIdx.x * blockDim.x + threadIdx.x;
  if (j < H) { const float v = h0[j]; hbuf[j] = v; cbuf[j] = 0.0f; out0[j] = v; }
}

// ---------------- Per-step GEMV: part[kb, col] = sum_{k in chunk} h[k]*Wsum[k,col] --
// grid = (CBLK, nChunks), block = 256. float4 per thread -> 1024 cols/block.
// STREAM_NT=true for the non-L2-resident K range (TH=NT loads).
template <bool FUSED, bool STREAM_NT>
__global__ __launch_bounds__(256) void gemv_kernel(
    const float* __restrict__ A,      // FUSED: Wsum ; else: W + E*4H (recurrent part)
    const float* __restrict__ B,      // FUSED: unused ; else: U
    const float* __restrict__ h,      // [H]
    float* __restrict__ part,         // [KSPLIT, 4H]
    int kbase)
{
  __shared__ float hs[KCH];
  const int tid = threadIdx.x;
  const int cb = blockIdx.x, kb = blockIdx.y + kbase;
  if (tid < KCH) hs[tid] = h[kb * KCH + tid];
  __syncthreads();

  const int col = cb * CPB + tid * 4;
  const float* Ab = A + (size_t)kb * KCH * FOURH + col;
  const float* Bb = FUSED ? nullptr : (B + (size_t)kb * KCH * FOURH + col);

  float ax = 0.f, ay = 0.f, az = 0.f, aw = 0.f;
  #pragma unroll 4
  for (int k = 0; k < KCH; ++k) {
    const float hv = hs[k];
    const v4f* wp = (const v4f*)(Ab + (size_t)k * FOURH);
    const v4f w = STREAM_NT ? __builtin_nontemporal_load(wp) : *wp;
    float wx = w.x, wy = w.y, wz = w.z, ww = w.w;
    if (!FUSED) {
      const v4f* up = (const v4f*)(Bb + (size_t)k * FOURH);
      const v4f u = STREAM_NT ? __builtin_nontemporal_load(up) : *up;
      wx += u.x; wy += u.y; wz += u.z; ww += u.w;
    }
    ax = fmaf(hv, wx, ax); ay = fmaf(hv, wy, ay);
    az = fmaf(hv, wz, az); aw = fmaf(hv, ww, aw);
  }
  v4f acc; acc.x = ax; acc.y = ay; acc.z = az; acc.w = aw;
  *(v4f*)(part + (size_t)kb * FOURH + col) = acc;
}

// ---------------- Gate kernel: reduce partials, LSTM update, emit orders row --------
__global__ __launch_bounds__(256) void gate_kernel(
    const float* __restrict__ Zxt,    // Zx + t*4H
    const float* __restrict__ part,   // [KSPLIT, 4H]
    float* __restrict__ hbuf, float* __restrict__ cbuf,
    float* __restrict__ out_row)      // orders row t+1 (or nullptr at last step)
{
  const int j = blockIdx.x * blockDim.x + threadIdx.x;   // 0..H-1
  float zi = Zxt[j], zf = Zxt[j + H], zg = Zxt[j + 2 * H], zo = Zxt[j + 3 * H];
  #pragma unroll
  for (int p = 0; p < KSPLIT; ++p) {
    const float* pr = part + (size_t)p * FOURH;
    zi += pr[j]; zf += pr[j + H]; zg += pr[j + 2 * H]; zo += pr[j + 3 * H];
  }
  const float si = 1.0f / (1.0f + expf(-zi));
  const float sf = 1.0f / (1.0f + expf(-zf));
  const float so = 1.0f / (1.0f + expf(-zo));
  const float c  = sf * cbuf[j] + si * zg;     // linear candidate (activation=None)
  const float hn = so * c;                     // linear output
  cbuf[j] = c; hbuf[j] = hn;
  if (out_row) out_row[j] = hn;
}

extern "C" void kernel_launch(void* const* d_in, const int* in_sizes, int n_in,
                              void* d_out, int out_size, void* d_ws, size_t ws_size,
                              hipStream_t stream) {
  const float* henc = (const float*)d_in[0];   // [81, 256]
  const float* h0   = (const float*)d_in[1];   // [4096]
  const float* W    = (const float*)d_in[2];   // [4352, 16384]
  const float* U    = (const float*)d_in[3];   // [4096, 16384]
  const float* bias = (const float*)d_in[4];   // [16384]
  float* out = (float*)d_out;                  // [81, 4096]
  float* ws  = (float*)d_ws;

  // Workspace layout (floats); all offsets 16B-aligned for float4 paths.
  const size_t offZx   = 0;                                  // 81*16384
  const size_t offPart = offZx + (size_t)P * FOURH;          // 32*16384
  const size_t offHb   = offPart + (size_t)KSPLIT * FOURH;   // 4096
  const size_t offCb   = offHb + H;                          // 4096
  const size_t offWsum = offCb + H;                          // 4096*16384 (optional)
  const size_t needFused = (offWsum + (size_t)H * FOURH) * sizeof(float);
  const bool fused = (ws_size >= needFused);                 // ~275 MB -> halves stream

  float* Zx   = ws + offZx;
  float* part = ws + offPart;
  float* hbuf = ws + offHb;
  float* cbuf = ws + offCb;
  float* Wsum = ws + offWsum;

  if (fused) {
    const long n4     = (long)((size_t)H * FOURH / 4);
    const long n4keep = (long)((size_t)KEEP_FUSED * KCH * FOURH / 4);
    wsum_kernel<<<2048, 256, 0, stream>>>(W, U, Wsum, n4, n4keep);
  }
  zx_wmma_kernel<<<dim3(FOURH / 16, (P + 15) / 16), 32, 0, stream>>>(henc, W, bias, Zx);
  init_kernel<<<H / 256, 256, 0, stream>>>(h0, hbuf, cbuf, out);

  const float* Wrec = W + (size_t)E * FOURH;
  for (int t = 0; t < P; ++t) {
    if (fused) {
      // Resident K-chunks (RT loads, served from L2 after step 0) ...
      gemv_kernel<true, false><<<dim3(CBLK, KEEP_FUSED), 256, 0, stream>>>(
          Wsum, nullptr, hbuf, part, 0);
      // ... streaming tail (TH=NT loads: never evicts the resident slice).
      gemv_kernel<true, true><<<dim3(CBLK, KSPLIT - KEEP_FUSED), 256, 0, stream>>>(
          Wsum, nullptr, hbuf, part, KEEP_FUSED);
    } else {
      gemv_kernel<false, false><<<dim3(CBLK, KEEP_FB), 256, 0, stream>>>(
          Wrec, U, hbuf, part, 0);
      gemv_kernel<false, true><<<dim3(CBLK, KSPLIT - KEEP_FB), 256, 0, stream>>>(
          Wrec, U, hbuf, part, KEEP_FB);
    }
    float* orow = (t + 1 < P) ? (out + (size_t)(t + 1) * H) : nullptr;
    gate_kernel<<<H / 256, 256, 0, stream>>>(Zx + (size_t)t * FOURH, part,
                                             hbuf, cbuf, orow);
  }
}